// LengthRegulator_22866405884035
// MI455X (gfx1250) — compile-verified
//
#include <hip/hip_runtime.h>
#include <hip/hip_bf16.h>

// LengthRegulator for MI455X (gfx1250, wave32).
// Phase 1: per-batch segment scan + searchsorted -> gather indices in d_ws.
// Phase 2: row gather. Primary path: gfx1250 async data mover
//          (GLOBAL_LOAD_ASYNC_TO_LDS_B128 -> s_wait_asynccnt ->
//           GLOBAL_STORE_ASYNC_FROM_LDS_B128), one wave32 per 2KB row.
//          Fallback: b128 VGPR loads + non-temporal b128 stores.

typedef __attribute__((ext_vector_type(4))) float v4f;
typedef __attribute__((ext_vector_type(4))) int   v4i;

// Builtin signature (from hipcc diagnostic): param0 = int4 in global AS(1),
// param1 = LDS pointer, then imm offset, imm cpol.
typedef __attribute__((address_space(1))) v4i* g_v4i_p;   // global int4*
typedef __attribute__((address_space(3))) v4i* l_v4i_p;   // LDS int4*

#if __has_builtin(__builtin_amdgcn_global_load_async_to_lds_b128) && \
    __has_builtin(__builtin_amdgcn_global_store_async_from_lds_b128) && \
    __has_builtin(__builtin_amdgcn_s_wait_asynccnt)
#define LR_USE_ASYNC_LDS 1
#else
#define LR_USE_ASYNC_LDS 0
#endif

#define SCAN_THREADS 256
#define SEG_CAP      2048   // supports 2*T up to 2048 (reference: 2*512 = 1024)
#define GATHER_BLOCK 256    // 8 waves -> 8 rows per block
#define ROW_FLOATS   512    // D for the async fast path (2 KB rows)

__global__ void lr_index_kernel(const int* __restrict__ phone,
                                const int* __restrict__ sil,
                                const int* __restrict__ src_lens,
                                int* __restrict__ src_idx,      // [B, max_len]
                                float* __restrict__ tgt_len,    // [B] (tail of d_out)
                                int T, int max_len)
{
    __shared__ int s_cum[SEG_CAP];
    __shared__ int s_part[SCAN_THREADS];

    const int b   = blockIdx.x;
    const int tid = threadIdx.x;
    const int nseg = 2 * T;                      // 1024 for reference shapes
    const int ept  = SEG_CAP / SCAN_THREADS;     // 8 elements per thread (padded)

    const int L = src_lens[b];
    const int* ph = phone + (size_t)b * T;
    const int* si = sil   + (size_t)b * T;

    // interleaved segment durations: even j -> phone, odd j -> silence
    for (int j = tid; j < SEG_CAP; j += SCAN_THREADS) {
        int d = 0;
        if (j < nseg) {
            int i = j >> 1;
            if ((j & 1) == 0) {
                int p = ph[i];
                d = (i < L) ? (p > 1 ? p : 1) : 1;
            } else {
                int s = si[i];
                d = (i < L) ? (s > 0 ? s : 0) : 0;
            }
        }
        s_cum[j] = d;
    }
    __syncthreads();

    // per-thread serial inclusive scan over its contiguous chunk
    const int base = tid * ept;
    int acc = 0;
    int local[8];
    #pragma unroll
    for (int k = 0; k < 8; ++k) {
        if (k < ept) {
            acc += s_cum[base + k];
            local[k] = acc;
        }
    }
    s_part[tid] = acc;
    __syncthreads();

    // Hillis-Steele inclusive scan over 256 chunk totals
    for (int off = 1; off < SCAN_THREADS; off <<= 1) {
        int v = s_part[tid];
        int u = (tid >= off) ? s_part[tid - off] : 0;
        __syncthreads();
        s_part[tid] = v + u;
        __syncthreads();
    }
    const int excl = (tid == 0) ? 0 : s_part[tid - 1];
    #pragma unroll
    for (int k = 0; k < 8; ++k) {
        if (k < ept) s_cum[base + k] = local[k] + excl;
    }
    __syncthreads();

    const int total = s_cum[nseg - 1];
    if (tid == 0) tgt_len[b] = (float)total;

    // searchsorted(cum, t, side='right'): first j with cum[j] > t
    for (int t = tid; t < max_len; t += SCAN_THREADS) {
        int lo = 0, hi = nseg;
        while (lo < hi) {
            int mid = (lo + hi) >> 1;
            if (s_cum[mid] > t) hi = mid; else lo = mid + 1;
        }
        int seg = (lo > nseg - 1) ? (nseg - 1) : lo;
        int src = ((seg & 1) == 0) ? (seg >> 1) : 0;   // odd segment -> silence row 0
        src_idx[(size_t)b * max_len + t] = (t < total) ? src : -1;
    }
}

// One wave32 per output row (wave-uniform gather index, no divergence).
__global__ void lr_gather_kernel(const float* __restrict__ x,
                                 const int* __restrict__ src_idx,
                                 float* __restrict__ out,
                                 int T, int D, int max_len, int nrows)
{
#if LR_USE_ASYNC_LDS
    __shared__ float s_stage[(GATHER_BLOCK / 32) * ROW_FLOATS];   // 2 KB per wave
#endif
    const int lane = threadIdx.x & 31;
    const int wave = threadIdx.x >> 5;
    const int row  = blockIdx.x * (blockDim.x >> 5) + wave;
    if (row >= nrows) return;

    const int b   = row / max_len;
    const int src = src_idx[row];                   // wave-uniform
    float* orow   = out + (size_t)row * D;

    if (src < 0) {
        // padding region: stream zeros, non-temporal (write-once output)
        v4f z = {0.0f, 0.0f, 0.0f, 0.0f};
        v4f* ov = (v4f*)orow;
        const int nchunk = D >> 7;
        for (int c = 0; c < nchunk; ++c)
            __builtin_nontemporal_store(z, ov + lane + 32 * c);
        return;
    }

    const float* xrow = x + ((size_t)b * T + src) * D;

#if LR_USE_ASYNC_LDS
    if (D == ROW_FLOATS) {
        // gfx1250 async data mover: HBM -> LDS -> HBM, no VGPR data movement.
        // Each instruction moves 16 B/lane = 512 B/wave; imm offset applies to
        // both global and LDS addresses (ISA 10.x async pseudocode).
        float* lrow = &s_stage[wave * ROW_FLOATS + lane * 4];
        l_v4i_p lp = (l_v4i_p)lrow;
        g_v4i_p gp = (g_v4i_p)(xrow + (size_t)lane * 4);
        __builtin_amdgcn_global_load_async_to_lds_b128(gp, lp, 0,    0);
        __builtin_amdgcn_global_load_async_to_lds_b128(gp, lp, 512,  0);
        __builtin_amdgcn_global_load_async_to_lds_b128(gp, lp, 1024, 0);
        __builtin_amdgcn_global_load_async_to_lds_b128(gp, lp, 1536, 0);
        __builtin_amdgcn_s_wait_asynccnt(0);        // row resident in LDS
        g_v4i_p op = (g_v4i_p)(orow + (size_t)lane * 4);
        __builtin_amdgcn_global_store_async_from_lds_b128(op, lp, 0,    0);
        __builtin_amdgcn_global_store_async_from_lds_b128(op, lp, 512,  0);
        __builtin_amdgcn_global_store_async_from_lds_b128(op, lp, 1024, 0);
        __builtin_amdgcn_global_store_async_from_lds_b128(op, lp, 1536, 0);
        __builtin_amdgcn_s_wait_asynccnt(0);        // LDS reusable / wave may exit
        return;
    }
#endif

    // Generic VGPR path: b128 loads, non-temporal b128 stores.
    const v4f* xv = (const v4f*)xrow;
    v4f* ov = (v4f*)orow;
    const int nchunk = D >> 7;                      // float4s per lane (D/128)
    __builtin_prefetch(xrow + lane * 4, 0, 3);      // gfx1250 global_prefetch_b8
    v4f r[8];
    for (int c = 0; c < nchunk; ++c)
        r[c & 7] = xv[lane + 32 * c];
    for (int c = 0; c < nchunk; ++c)
        __builtin_nontemporal_store(r[c & 7], ov + lane + 32 * c);
}

extern "C" void kernel_launch(void* const* d_in, const int* in_sizes, int n_in,
                              void* d_out, int out_size, void* d_ws, size_t ws_size,
                              hipStream_t stream) {
    const float* x        = (const float*)d_in[0];   // [B, T, D] fp32
    const int*   phone    = (const int*)d_in[1];     // [B, T] int32
    const int*   sil      = (const int*)d_in[2];     // [B, T] int32
    const int*   src_lens = (const int*)d_in[3];     // [B] int32
    // d_in[4] is max_len on device; derive it from sizes instead (graph-capture safe).

    const int B = in_sizes[3];
    const int T = in_sizes[1] / B;
    const int D = in_sizes[0] / in_sizes[1];
    const int max_len = (out_size - B) / (B * D);    // out = B*max_len*D floats + B lengths

    float* out      = (float*)d_out;
    float* tgt_tail = out + (size_t)B * max_len * D;
    int*   src_idx  = (int*)d_ws;                    // B*max_len ints (256 KB)

    lr_index_kernel<<<B, SCAN_THREADS, 0, stream>>>(phone, sil, src_lens,
                                                    src_idx, tgt_tail, T, max_len);

    const int nrows = B * max_len;
    const int rows_per_block = GATHER_BLOCK / 32;    // one row per wave32
    const int blocks = (nrows + rows_per_block - 1) / rows_per_block;
    lr_gather_kernel<<<blocks, GATHER_BLOCK, 0, stream>>>(x, src_idx, out, T, D, max_len, nrows);
}